// GraphConvolutionNetworkConv_51135880626289
// MI455X (gfx1250) — compile-verified
//
#include <hip/hip_runtime.h>

typedef __attribute__((ext_vector_type(2))) float v2f;
typedef __attribute__((ext_vector_type(8))) float v8f;

#define N_NODES 50000
#define N_EDGES 800000
#define CH 128          // IN_CH == OUT_CH == 128

// ---------------------------------------------------------------------------
// 1) zero the degree buffer
__global__ void gcn_zero_deg(float* __restrict__ deg) {
    int i = blockIdx.x * blockDim.x + threadIdx.x;
    if (i < N_NODES) deg[i] = 0.0f;
}

// 2) out[j][c] = b[c]   (scatter-add target must start at bias)
__global__ void gcn_init_out(float* __restrict__ out, const float* __restrict__ b) {
    int i = blockIdx.x * blockDim.x + threadIdx.x;
    if (i < N_NODES * CH) out[i] = b[i & (CH - 1)];
}

// 3) deg[dst] += 1 per edge
__global__ void gcn_degree(const int* __restrict__ ei, float* __restrict__ deg) {
    int e = blockIdx.x * blockDim.x + threadIdx.x;
    if (e < N_EDGES) atomicAdd(&deg[ei[N_EDGES + e]], 1.0f);
}

// 4) dinv[i] = deg[i] > 0 ? rsqrt(deg[i]) : 0
__global__ void gcn_dinv(const float* __restrict__ deg, float* __restrict__ dinv) {
    int i = blockIdx.x * blockDim.x + threadIdx.x;
    if (i < N_NODES) {
        float d = deg[i];
        dinv[i] = (d > 0.0f) ? rsqrtf(d) : 0.0f;
    }
}

// ---------------------------------------------------------------------------
// 5) XW = X @ W via V_WMMA_F32_16X16X4_F32 (f32 matrix cores, full precision)
//    Block = 256 threads = 8 waves. Wave w computes the 16x16 tile at
//    (m0 = 16*blockIdx.x, n0 = 16*w). K = 128 -> 32 WMMAs per tile.
//    W (128x128 f32 = 64 KB) is staged once per block in LDS.
__global__ void __launch_bounds__(256, 2)
gcn_gemm_wmma(const float* __restrict__ X, const float* __restrict__ W,
              float* __restrict__ XW) {
    __shared__ float Wlds[CH * CH];   // 64 KB of the 320 KB WGP LDS

    // cooperative float4 load of W into LDS
    for (int i = threadIdx.x; i < (CH * CH) / 4; i += 256) {
        ((float4*)Wlds)[i] = ((const float4*)W)[i];
    }
    __syncthreads();

    const int wave = threadIdx.x >> 5;
    const int lane = threadIdx.x & 31;
    const int half = lane >> 4;       // 0: lanes 0-15, 1: lanes 16-31
    const int l15  = lane & 15;

    const int m0 = blockIdx.x * 16;   // 3125 * 16 == 50000 exactly, no tail
    const int n0 = wave * 16;         // 8 waves cover N = 128 exactly
    const int n  = n0 + l15;

    const float* __restrict__ Xrow = X + (size_t)(m0 + l15) * CH;

    v8f c = {};
    #pragma unroll
    for (int k = 0; k < CH; k += 4) {
        // A fragment (16x4 f32): lanes 0-15 hold K={k,k+1}, lanes 16-31 K={k+2,k+3}
        const int ka = k + (half << 1);
        v2f a;
        a.x = Xrow[ka];
        a.y = Xrow[ka + 1];
        // B fragment (4x16 f32): row-striped across lanes, mirrored K split
        v2f bm;
        bm.x = Wlds[ka * CH + n];
        bm.y = Wlds[(ka + 1) * CH + n];
        // (neg_a, A, neg_b, B, c_mod, C, reuse_a, reuse_b)
        c = __builtin_amdgcn_wmma_f32_16x16x4_f32(false, a, false, bm,
                                                  (short)0, c, false, false);
    }

    // C/D layout: VGPR r -> lanes 0-15: (M = r, N = lane), lanes 16-31: (M = r+8)
    #pragma unroll
    for (int r = 0; r < 8; ++r) {
        XW[(size_t)(m0 + r + half * 8) * CH + n] = c[r];
    }
}

// ---------------------------------------------------------------------------
// 6) out[dst] += dinv[src]*dinv[dst] * XW[src]
//    One wave per edge; each lane owns 4 channels (float4 coalesced read,
//    4x global_atomic_add_f32). 512 B per edge read, fully coalesced.
__global__ void __launch_bounds__(256)
gcn_scatter(const int* __restrict__ ei, const float* __restrict__ dinv,
            const float* __restrict__ xw, float* __restrict__ out) {
    const int gwave = (blockIdx.x * blockDim.x + threadIdx.x) >> 5;
    const int lane  = threadIdx.x & 31;
    if (gwave >= N_EDGES) return;

    const int src = ei[gwave];
    const int dst = ei[N_EDGES + gwave];
    const float norm = dinv[src] * dinv[dst];

    const float4 v = ((const float4*)(xw + (size_t)src * CH))[lane];
    float* o = out + (size_t)dst * CH + lane * 4;
    atomicAdd(o + 0, v.x * norm);
    atomicAdd(o + 1, v.y * norm);
    atomicAdd(o + 2, v.z * norm);
    atomicAdd(o + 3, v.w * norm);
}

// ---------------------------------------------------------------------------
extern "C" void kernel_launch(void* const* d_in, const int* in_sizes, int n_in,
                              void* d_out, int out_size, void* d_ws, size_t ws_size,
                              hipStream_t stream) {
    const float* X  = (const float*)d_in[0];   // [50000,128]
    const int*   ei = (const int*)d_in[1];     // [2,800000]
    const float* W  = (const float*)d_in[2];   // [128,128]
    const float* b  = (const float*)d_in[3];   // [128]
    float* out = (float*)d_out;                // [50000,128]

    // workspace layout
    float* deg  = (float*)d_ws;                               // 200 KB
    float* dinv = deg + N_NODES;                              // 200 KB
    float* xw   = (float*)((char*)d_ws + (size_t)512 * 1024); // 25.6 MB

    const int T = 256;

    gcn_zero_deg<<<(N_NODES + T - 1) / T, T, 0, stream>>>(deg);
    gcn_init_out<<<(N_NODES * CH + T - 1) / T, T, 0, stream>>>(out, b);
    gcn_degree<<<(N_EDGES + T - 1) / T, T, 0, stream>>>(ei, deg);
    gcn_dinv<<<(N_NODES + T - 1) / T, T, 0, stream>>>(deg, dinv);

    // GEMM: 3125 M-tiles, 8 waves/block cover all 8 N-tiles
    gcn_gemm_wmma<<<N_NODES / 16, T, 0, stream>>>(X, W, xw);

    // one wave per edge: 800000 waves / 8 waves-per-block
    gcn_scatter<<<N_EDGES / 8, T, 0, stream>>>(ei, dinv, xw, out);
}